// DeformConv_28647431864973
// MI455X (gfx1250) — compile-verified
//
#include <hip/hip_runtime.h>

typedef __attribute__((ext_vector_type(2))) float v2f;
typedef __attribute__((ext_vector_type(8))) float v8f;

#define Hc   128
#define Wc   128
#define Cc   64
#define Oc   64
#define KKc  9
#define CK   576          // Cc * KKc, GEMM K dimension
#define TILE 32           // pixels per workgroup
#define PROW 580          // padded LDS pixel-row stride (floats); 580 % 64 == 4
                          // -> half-wave b64 reads hit disjoint bank classes

// ---------------------------------------------------------------------------
// Pre-pass: reorder weight[o][c][k] -> WMMA A-fragment layout in d_ws.
// Flat float index: (((s*4 + m)*32) + lane)*2 + j
//   s    = K-step (ck/4), m = o-tile, lane = wave lane, j = VGPR (0/1)
//   A element = W[o = m*16 + (lane&15)][ck = 4*s + 2*(lane>>4) + j]
//   ck order is k*64 + c  (k-major, c-minor), matching the LDS S columns
// ---------------------------------------------------------------------------
__global__ void dcn_weight_reorder(const float* __restrict__ w,
                                   float* __restrict__ wr) {
    int idx = blockIdx.x * 256 + threadIdx.x;
    if (idx >= (CK / 4) * 4 * 32 * 2) return;   // 36864 floats
    int j = idx & 1;
    int l = (idx >> 1) & 31;
    int m = (idx >> 6) & 3;
    int s = idx >> 8;
    int ck = 4 * s + ((l >> 4) << 1) + j;
    int k  = ck >> 6;        // ck / 64
    int c  = ck & 63;        // ck % 64
    int o  = m * 16 + (l & 15);
    wr[idx] = w[(o * Cc + c) * KKc + k];
}

// ---------------------------------------------------------------------------
// Fused deformable-conv kernel: 128 threads (4 waves), 32-pixel row tile.
//   Stage 1: bilinear im2col into LDS, transposed:  S[pixel][ck]
//   Stage 2: out[64 x 32] = A[64 x 576] * S[576 x 32] via v_wmma_f32_16x16x4_f32
// ---------------------------------------------------------------------------
__launch_bounds__(128)
__global__ void dcn_fused(const float* __restrict__ data,
                          const float* __restrict__ offs,
                          const float* __restrict__ wr,
                          float* __restrict__ out) {
    extern __shared__ float smem[];              // TILE * PROW floats (74240 B)

    int blk = blockIdx.x;                        // 4096 = 8 * 128 * 4
    int wt  = blk & 3;
    int h   = (blk >> 2) & 127;
    int b   = blk >> 9;
    int w0  = wt * TILE;

    int tid = threadIdx.x;
    int p   = tid & 31;                          // pixel within tile
    int cg  = tid >> 5;                          // channel group 0..3
    int wpix = w0 + p;

    const float* dbase = data + (size_t)b * Cc * Hc * Wc;
    const float* obase = offs + (size_t)b * 18 * Hc * Wc;
    float* srow = smem + p * PROW;

    // ---- Stage 1: deformable bilinear sampling (im2col into LDS) ----
    for (int k = 0; k < KKc; ++k) {
        int ki = k / 3, kj = k % 3;
        float off_y = obase[((size_t)(2 * k)     * Hc + h) * Wc + wpix];
        float off_x = obase[((size_t)(2 * k + 1) * Hc + h) * Wc + wpix];
        float py = off_y + (float)(h - 1 + ki);
        float px = off_x + (float)(wpix - 1 + kj);

        float y0f = floorf(py), x0f = floorf(px);
        int   y0  = (int)y0f,   x0  = (int)x0f;
        int   y1  = y0 + 1,     x1  = x0 + 1;
        float wy1 = py - y0f,   wx1 = px - x0f;
        float wy0 = 1.0f - wy1, wx0 = 1.0f - wx1;

        bool vy0 = (y0 >= 0) && (y0 < Hc);
        bool vy1 = (y1 >= 0) && (y1 < Hc);
        bool vx0 = (x0 >= 0) && (x0 < Wc);
        bool vx1 = (x1 >= 0) && (x1 < Wc);
        int y0c = min(max(y0, 0), Hc - 1), y1c = min(max(y1, 0), Hc - 1);
        int x0c = min(max(x0, 0), Wc - 1), x1c = min(max(x1, 0), Wc - 1);

        float w00 = wy0 * wx0 * ((vy0 && vx0) ? 1.0f : 0.0f);
        float w01 = wy0 * wx1 * ((vy0 && vx1) ? 1.0f : 0.0f);
        float w10 = wy1 * wx0 * ((vy1 && vx0) ? 1.0f : 0.0f);
        float w11 = wy1 * wx1 * ((vy1 && vx1) ? 1.0f : 0.0f);

        int i00 = y0c * Wc + x0c, i01 = y0c * Wc + x1c;
        int i10 = y1c * Wc + x0c, i11 = y1c * Wc + x1c;

        // 4 corners + weights shared across all 64 channels
        for (int c = cg; c < Cc; c += 4) {
            const float* cb = dbase + (size_t)c * Hc * Wc;
            float sv = w00 * cb[i00] + w01 * cb[i01]
                     + w10 * cb[i10] + w11 * cb[i11];
            srow[k * Cc + c] = sv;               // transposed: ck contiguous
        }
    }
    __syncthreads();

    // ---- Stage 2: WMMA GEMM (fp32 exact) ----
    int lane = tid & 31;
    int m    = tid >> 5;                         // wave id = o-tile (0..3)
    int pl   = lane & 15;                        // N within 16
    int kof  = (lane >> 4) << 1;                 // K pair base: 0 or 2

    const float* b0p = smem + (size_t)pl * PROW + kof;         // tile 0 pixels
    const float* b1p = smem + (size_t)(pl + 16) * PROW + kof;  // tile 1 pixels

    v8f acc0 = {};
    v8f acc1 = {};
    const v2f* wr2 = (const v2f*)wr;

    for (int s = 0; s < CK / 4; ++s) {
        v2f a  = wr2[(s * 4 + m) * 32 + lane];   // A fragment (global, L2-hot)
        v2f b0 = *(const v2f*)(b0p + 4 * s);     // one aligned ds_load_b64
        v2f b1 = *(const v2f*)(b1p + 4 * s);
        acc0 = __builtin_amdgcn_wmma_f32_16x16x4_f32(
                   false, a, false, b0, (short)0, acc0, false, false);
        acc1 = __builtin_amdgcn_wmma_f32_16x16x4_f32(
                   false, a, false, b1, (short)0, acc1, false, false);
    }

    // ---- Store: D layout -> NCHW ----
    // VGPR r: lanes 0-15 = (M=r, N=lane), lanes 16-31 = (M=r+8, N=lane-16)
    float* ob = out + ((size_t)b * Oc * Hc + (size_t)h) * Wc;
    int olocal = (lane >> 4) * 8;
    int wcol   = w0 + (lane & 15);
#pragma unroll
    for (int r = 0; r < 8; ++r) {
        int o = m * 16 + olocal + r;
        ob[(size_t)o * Hc * Wc + wcol]      = acc0[r];
        ob[(size_t)o * Hc * Wc + wcol + 16] = acc1[r];
    }
}

// ---------------------------------------------------------------------------
extern "C" void kernel_launch(void* const* d_in, const int* in_sizes, int n_in,
                              void* d_out, int out_size, void* d_ws, size_t ws_size,
                              hipStream_t stream) {
    const float* data   = (const float*)d_in[0];   // (8,64,128,128)
    const float* offset = (const float*)d_in[1];   // (8,18,128,128)
    const float* weight = (const float*)d_in[2];   // (64,64,3,3)
    float*       out    = (float*)d_out;           // (8,64,128,128)
    float*       wr     = (float*)d_ws;            // 147456 B reordered weights

    int nWr = (CK / 4) * 4 * 32 * 2;               // 36864
    dcn_weight_reorder<<<(nWr + 255) / 256, 256, 0, stream>>>(weight, wr);

    size_t ldsBytes = (size_t)TILE * PROW * sizeof(float);   // 74240 B
    dcn_fused<<<4096, 128, ldsBytes, stream>>>(data, offset, wr, out);
}